// MultiGRU_21157008900485
// MI455X (gfx1250) — compile-verified
//
#include <hip/hip_runtime.h>
#include <cmath>

// ---------------------------------------------------------------------------
// MultiGRU (batch-1, 3 stacked TF-style GRU cells), fp32.
// GEMVs done with V_WMMA_F32_16X16X4_F32: A = input vector broadcast into all
// 16 rows (2 VGPRs/lane), B = 4x16 fp32 weight tile (coalesced row loads),
// D row 0 = the 16 partial dot products. Split-K via global_atomic_add_f32.
//
// Roofline: ~98 MB of fp32 weights, ~49 MFLOP -> 0.5 FLOP/byte, pure
// bandwidth-bound. First pass floor ~4.2 us at 23.3 TB/s HBM; on graph
// replays the 98 MB weight set is resident in the 192 MB L2, so we keep the
// default (RT) temporal hint instead of NT to exploit L2 residency.
// ---------------------------------------------------------------------------

typedef __attribute__((ext_vector_type(2))) float v2f;
typedef __attribute__((ext_vector_type(8))) float v8f;

#define KCHUNK 256  // K elements per wave (64 WMMA k-steps of 4)

__device__ __forceinline__ void atom_add_f32(float* p, float v) {
  __hip_atomic_fetch_add(p, v, __ATOMIC_RELAXED, __HIP_MEMORY_SCOPE_AGENT);
}

// One wave (32 threads) per block. blockIdx.x = N-tile (16 outputs),
// blockIdx.y = K-chunk. acc_out[n] += sum_{k in chunk} v[k]*W[k*N+n].
__global__ void __launch_bounds__(32)
k_gemv_wmma(const float* __restrict__ v, const float* __restrict__ W,
            float* __restrict__ acc_out, int K, int N) {
  const int lane = threadIdx.x;       // 0..31
  const int half = lane >> 4;         // 0: K=0,1  1: K=2,3  (32-bit A/B layout)
  const int col  = lane & 15;         // N within tile
  const int n    = blockIdx.x * 16 + col;
  const int kbeg = blockIdx.y * KCHUNK;
  int kend = kbeg + KCHUNK;
  if (kend > K) kend = K;             // uniform per wave; K always multiple of 4

  v8f acc = {};                       // 16x16 f32 C/D = 8 VGPRs
#pragma unroll 4
  for (int k0 = kbeg; k0 < kend; k0 += 4) {
    const int kr = k0 + 2 * half;
    // A 16x4: every row = same 4 vector elements (broadcast) -> D rows equal.
    v2f a, b;
    a.x = v[kr];                      // contiguous pair -> global_load_b64
    a.y = v[kr + 1];
    // B 4x16: lane col holds W[kr][n] (VGPR0) and W[kr+1][n] (VGPR1).
    const float* wp = W + (size_t)kr * (size_t)N + n;
    b.x = wp[0];
    b.y = wp[(size_t)N];
    __builtin_prefetch(wp + 8 * (size_t)N, 0, 1);  // global_prefetch_b8
    // 8 args: (neg_a, A, neg_b, B, c_mod, C, reuse_a, reuse_b)
    acc = __builtin_amdgcn_wmma_f32_16x16x4_f32(
        false, a, false, b, (short)0, acc, false, false);
  }
  // D layout: VGPR0, lanes 0-15 -> M=0, N=lane  (all rows identical here)
  if (lane < 16) atom_add_f32(acc_out + n, acc[0]);
}

// Build vg = concat(x, h); zero the gate and candidate accumulators.
__global__ void k_prep(const float* __restrict__ x, const float* __restrict__ h,
                       float* __restrict__ vg, float* __restrict__ gacc,
                       float* __restrict__ cacc, int fi, int fo) {
  int i = blockIdx.x * blockDim.x + threadIdx.x;
  if (i < fi) vg[i] = x[i];
  if (i < fo) vg[fi + i] = h[i];
  if (i < 2 * fo) gacc[i] = 0.0f;
  if (i < fo) cacc[i] = 0.0f;
}

// gates = sigmoid(gacc + bg); rh = r*h; stash z.
__global__ void k_gates(const float* __restrict__ gacc, const float* __restrict__ bg,
                        const float* __restrict__ h, float* __restrict__ rh,
                        float* __restrict__ zb, int fo) {
  int i = blockIdx.x * blockDim.x + threadIdx.x;
  if (i >= fo) return;
  float r = 1.0f / (1.0f + expf(-(gacc[i] + bg[i])));
  float z = 1.0f / (1.0f + expf(-(gacc[fo + i] + bg[fo + i])));
  rh[i] = r * h[i];
  zb[i] = z;
}

// n = z*h + (1-z)*tanh(cacc + bc); optionally mirror to a second output slot.
__global__ void k_out(const float* __restrict__ cacc, const float* __restrict__ bc,
                      const float* __restrict__ zb, const float* __restrict__ h,
                      float* __restrict__ o1, float* __restrict__ o2, int fo) {
  int i = blockIdx.x * blockDim.x + threadIdx.x;
  if (i >= fo) return;
  float c = tanhf(cacc[i] + bc[i]);
  float z = zb[i];
  float n = z * h[i] + (1.0f - z) * c;
  o1[i] = n;
  if (o2) o2[i] = n;
}

extern "C" void kernel_launch(void* const* d_in, const int* in_sizes, int n_in,
                              void* d_out, int out_size, void* d_ws, size_t ws_size,
                              hipStream_t stream) {
  (void)in_sizes; (void)n_in; (void)out_size; (void)ws_size;
  const float* x0 = (const float*)d_in[0];
  const float* hs[3] = {(const float*)d_in[1], (const float*)d_in[2],
                        (const float*)d_in[3]};
  float* out = (float*)d_out;   // layout: n2[2048] n0[512] n1[1024] n2[2048]
  float* ws  = (float*)d_ws;
  float* vg   = ws;             // 4096 floats (max K = 3072)
  float* gacc = ws + 4096;      // 4096 floats (max 2*fo)
  float* cacc = ws + 8192;      // 2048 floats
  float* rh   = ws + 10240;     // 2048 floats
  float* zb   = ws + 12288;     // 2048 floats

  const int fis[3] = {40, 512, 1024};
  const int fos[3] = {512, 1024, 2048};
  float* nout[3] = {out + 2048, out + 2560, out + 3584};

  const float* xcur = x0;
  for (int L = 0; L < 3; ++L) {
    const float* Wg = (const float*)d_in[4 + 4 * L];
    const float* bg = (const float*)d_in[5 + 4 * L];
    const float* Wc = (const float*)d_in[6 + 4 * L];
    const float* bc = (const float*)d_in[7 + 4 * L];
    const float* h  = hs[L];
    const int fi = fis[L], fo = fos[L], K = fi + fo;

    // 1) concat(x,h) + zero accumulators
    int npre = 2 * fo;
    k_prep<<<(npre + 255) / 256, 256, 0, stream>>>(xcur, h, vg, gacc, cacc, fi, fo);

    // 2) gates GEMV: [1,K] x [K, 2fo]
    dim3 gg((2 * fo) / 16, (K + KCHUNK - 1) / KCHUNK);
    k_gemv_wmma<<<gg, 32, 0, stream>>>(vg, Wg, gacc, K, 2 * fo);

    // 3) candidate GEMV, x-part: rows [0, fi) of Wc (only needs x)
    dim3 gx(fo / 16, (fi + KCHUNK - 1) / KCHUNK);
    k_gemv_wmma<<<gx, 32, 0, stream>>>(vg, Wc, cacc, fi, fo);

    // 4) sigmoid gates, build r*h, stash z
    k_gates<<<(fo + 255) / 256, 256, 0, stream>>>(gacc, bg, h, rh, zb, fo);

    // 5) candidate GEMV, h-part: rows [fi, fi+fo) of Wc against r*h
    dim3 gh(fo / 16, (fo + KCHUNK - 1) / KCHUNK);
    k_gemv_wmma<<<gh, 32, 0, stream>>>(rh, Wc + (size_t)fi * (size_t)fo, cacc, fo, fo);

    // 6) blend + write layer output (layer 2 also mirrored to d_out[0:2048))
    float* o2 = (L == 2) ? out : nullptr;
    k_out<<<(fo + 255) / 256, 256, 0, stream>>>(cacc, bc, zb, h, nout[L], o2, fo);

    xcur = nout[L];
  }
}